// VectorQuantizer_EMA_22110491640494
// MI455X (gfx1250) — compile-verified
//
#include <hip/hip_runtime.h>

#define N_ROWS 131072
#define DIM    128
#define KCODES 1024
#define EMA    0.85f
#define ONE_M_EMA 0.15f

#define LROW   132                    // 128 floats + 4 pad (bank-conflict-free ds_load_b64)
#define LBUF_F (16 * LROW)            // floats per LDS tile buffer

typedef __attribute__((ext_vector_type(2))) float v2f;
typedef __attribute__((ext_vector_type(8))) float v8f;
typedef __attribute__((ext_vector_type(4))) unsigned int v4u;
typedef __attribute__((ext_vector_type(8))) int v8i;
typedef __attribute__((ext_vector_type(4))) int v4i;

// ---------------- workspace layout (floats) ----------------
#define WS_ET     0
#define WS_ENORM  (KCODES*DIM)
#define WS_COUNTS (KCODES*DIM + KCODES)
#define WS_SUMS   (KCODES*DIM + 2*KCODES)
#define WS_LOSS   (KCODES*DIM + 2*KCODES + DIM*KCODES)

// ---------------- kernel 0a: transpose E -> ET, zero accumulators ----------------
__global__ void vq_init_kernel(const float* __restrict__ E,
                               float* __restrict__ ET,
                               float* __restrict__ counts,
                               float* __restrict__ sums,
                               float* __restrict__ lossAcc) {
    int t = blockIdx.x * blockDim.x + threadIdx.x;      // 0 .. D*K-1
    if (t < DIM * KCODES) {
        int k = t >> 7;          // t = k*D + d
        int d = t & (DIM - 1);
        ET[t] = E[d * KCODES + k];
        sums[t] = 0.0f;
    }
    if (t < KCODES) counts[t] = 0.0f;
    if (t == 0) *lossAcc = 0.0f;
}

// ---------------- kernel 0b: per-code squared norms ----------------
__global__ void vq_enorm_kernel(const float* __restrict__ ET,
                                float* __restrict__ enorm) {
    int k = blockIdx.x * blockDim.x + threadIdx.x;
    if (k < KCODES) {
        const float* row = ET + (size_t)k * DIM;
        float s = 0.0f;
        #pragma unroll 8
        for (int d = 0; d < DIM; ++d) s += row[d] * row[d];
        enorm[k] = s;
    }
}

// ---------------- TDM: stage one 16x128 fp32 code tile into LDS ----------------
// D# group0: count=1 | lds_addr | global_addr(57b) | type=2
// D# group1: data_size=4B, pad_enable, pad_interval=128 DWORDs (code 6),
//            pad_amount=4 DWORDs (code 3)  -> LDS row stride 132 floats,
//            tensor_dim0=128, tensor_dim1=1024, tile_dim0=128, tile_dim1=16,
//            tensor_dim0_stride=128
__device__ __forceinline__ void tdm_stage_tile(const float* gsrc, unsigned int lds_byte_addr) {
    unsigned long long ga = (unsigned long long)(uintptr_t)gsrc;
    v4u g0;
    g0.x = 1u;                                            // count = 1
    g0.y = lds_byte_addr;                                 // lds_addr
    g0.z = (unsigned int)ga;                              // global_addr[31:0]
    g0.w = ((unsigned int)(ga >> 32) & 0x01FFFFFFu)       // global_addr[56:32]
           | (2u << 30);                                  // type = 2 (image)
    v8i g1;
    g1[0] = (int)((2u << 16)      // data_size = 4 bytes
                | (1u << 20)      // pad_enable
                | (6u << 22)      // pad_interval: 128 DWORDs
                | (3u << 25));    // pad_amount: 4 DWORDs
    g1[1] = (int)(128u << 16);    // tensor_dim0 = 128  (bits 79:48 lo16)
    g1[2] = (int)(1024u << 16);   // tensor_dim0 hi = 0, tensor_dim1 lo16 = 1024
    g1[3] = (int)(128u << 16);    // tensor_dim1 hi = 0, tile_dim0 = 128
    g1[4] = 16;                   // tile_dim1 = 16, tile_dim2 = 0
    g1[5] = 128;                  // tensor_dim0_stride lo32 = 128
    g1[6] = 0;
    g1[7] = 0;
    v4i gz = {0, 0, 0, 0};
#if defined(__clang_major__) && (__clang_major__ >= 23)
    v8i gz8 = {0, 0, 0, 0, 0, 0, 0, 0};
    __builtin_amdgcn_tensor_load_to_lds(g0, g1, gz, gz, gz8, 0);
#else
    __builtin_amdgcn_tensor_load_to_lds(g0, g1, gz, gz, 0);
#endif
}

// ---------------- kernel 1: TDM-staged WMMA distances + argmin ----------------
// 8 waves/block; each wave owns a 16-row A tile; block shares LDS-staged B tiles.
__global__ void __launch_bounds__(256)
vq_main_kernel(const float* __restrict__ x,
               const float* __restrict__ ET,
               const float* __restrict__ enorm,
               float* __restrict__ q_out,
               float* __restrict__ counts,
               float* __restrict__ sums,
               float* __restrict__ lossAcc) {
    __shared__ float lbuf[2][LBUF_F];

    const int lane = threadIdx.x & 31;
    const int wave = threadIdx.x >> 5;
    const int tile = blockIdx.x * (blockDim.x >> 5) + wave;
    const int m0   = tile * 16;
    const int half = lane >> 4;
    const int l16  = lane & 15;
    const bool w0  = (wave == 0);   // wave-uniform: TDM issuer

    // ---- preload A tile: 16x128 fp32 in WMMA 16x4-chunk layout ----
    v2f a[32];
    float xnorm = 0.0f;
    const float* xrow = x + (size_t)(m0 + l16) * DIM;
    #pragma unroll
    for (int kk = 0; kk < 32; ++kk) {
        a[kk] = *(const v2f*)(xrow + kk * 4 + 2 * half);
        xnorm += a[kk].x * a[kk].x + a[kk].y * a[kk].y;
    }
    xnorm += __shfl_xor(xnorm, 16, 32);
    float rn[8];
    #pragma unroll
    for (int r = 0; r < 8; ++r)
        rn[r] = __shfl(xnorm, r + 8 * half, 32);

    float minv[8];
    int   mini[8];
    #pragma unroll
    for (int r = 0; r < 8; ++r) { minv[r] = 3.4e38f; mini[r] = 0; }

    // ---- prologue: stage first code tile ----
    if (w0) tdm_stage_tile(ET, (unsigned int)(uintptr_t)&lbuf[0][0]);

    // ---- sweep 64 column tiles of 16 codes, double-buffered via TDM ----
    for (int t = 0; t < KCODES / 16; ++t) {
        const int n0 = t * 16;
        if (w0) __builtin_amdgcn_s_wait_tensorcnt(0);
        __syncthreads();                       // tile t staged; buffer t^1 free
        if (w0 && (t + 1) < KCODES / 16)
            tdm_stage_tile(ET + (size_t)(n0 + 16) * DIM,
                           (unsigned int)(uintptr_t)&lbuf[(t + 1) & 1][0]);

        const float* bl = &lbuf[t & 1][l16 * LROW + 2 * half];
        v8f acc = {};
        #pragma unroll
        for (int kk = 0; kk < 32; ++kk) {
            v2f b = *(const v2f*)(bl + kk * 4);        // ds_load_b64, conflict-free
            acc = __builtin_amdgcn_wmma_f32_16x16x4_f32(
                false, a[kk], false, b, (short)0, acc, false, false);
        }
        float en = enorm[n0 + l16];
        #pragma unroll
        for (int r = 0; r < 8; ++r) {
            float dist = rn[r] + en - 2.0f * acc[r];
            if (dist < minv[r]) { minv[r] = dist; mini[r] = n0 + l16; }
        }
    }

    // ---- argmin butterfly within each 16-lane half (ties -> lowest index) ----
    #pragma unroll
    for (int off = 8; off >= 1; off >>= 1) {
        #pragma unroll
        for (int r = 0; r < 8; ++r) {
            float ov = __shfl_xor(minv[r], off, 32);
            int   oi = __shfl_xor(mini[r], off, 32);
            if (ov < minv[r] || (ov == minv[r] && oi < mini[r])) {
                minv[r] = ov; mini[r] = oi;
            }
        }
    }

    // ---- gather quantized rows, scatter EMA stats, loss partial ----
    float lloss = 0.0f;
    #pragma unroll
    for (int r = 0; r < 8; ++r) {
        const int row = m0 + r + 8 * half;
        const int c   = mini[r];
        if (l16 == 0) atomicAdd(&counts[c], 1.0f);
        const float* eq = ET + (size_t)c * DIM;
        const float* xr = x + (size_t)row * DIM;
        float* qo = q_out + (size_t)row * DIM;
        #pragma unroll
        for (int dd = 0; dd < 8; ++dd) {
            int d = l16 + dd * 16;
            float qv = eq[d];
            float xv = xr[d];
            qo[d] = qv;
            float df = qv - xv;
            lloss += df * df;
            atomicAdd(&sums[(size_t)d * KCODES + c], xv);
        }
    }
    #pragma unroll
    for (int off = 16; off >= 1; off >>= 1) lloss += __shfl_xor(lloss, off, 32);
    if (lane == 0) atomicAdd(lossAcc, lloss);
}

// ---------------- kernel 2: EMA blend + outputs ----------------
__global__ void vq_finalize_kernel(const float* __restrict__ ema_count,
                                   const float* __restrict__ emb_sum_old,
                                   const float* __restrict__ counts,
                                   const float* __restrict__ sums,
                                   const float* __restrict__ lossAcc,
                                   float* __restrict__ out_emb,
                                   float* __restrict__ out_cnt,
                                   float* __restrict__ out_sum,
                                   float* __restrict__ out_loss) {
    int t = blockIdx.x * blockDim.x + threadIdx.x;      // t = d*K + k
    if (t < DIM * KCODES) {
        int k = t & (KCODES - 1);
        float cn = ONE_M_EMA * ema_count[k] + EMA * counts[k];
        float sn = ONE_M_EMA * emb_sum_old[t] + EMA * sums[t];
        out_sum[t] = sn;
        out_emb[t] = sn / fmaxf(cn, 1e-5f);
        if (t < KCODES) out_cnt[t] = cn;
        if (t == 0)
            out_loss[0] = lossAcc[0] / (float)((long long)N_ROWS * DIM);
    }
}

extern "C" void kernel_launch(void* const* d_in, const int* in_sizes, int n_in,
                              void* d_out, int out_size, void* d_ws, size_t ws_size,
                              hipStream_t stream) {
    const float* x        = (const float*)d_in[0];   // [N, D]
    const float* emb      = (const float*)d_in[1];   // [D, K]
    const float* ema_cnt  = (const float*)d_in[2];   // [K]
    const float* emb_sum  = (const float*)d_in[3];   // [D, K]

    float* ws = (float*)d_ws;
    float* ET      = ws + WS_ET;
    float* enorm   = ws + WS_ENORM;
    float* counts  = ws + WS_COUNTS;
    float* sums    = ws + WS_SUMS;
    float* lossAcc = ws + WS_LOSS;

    float* out      = (float*)d_out;
    float* out_q    = out;                                    // N*D
    float* out_emb  = out_q   + (size_t)N_ROWS * DIM;         // D*K
    float* out_cnt  = out_emb + (size_t)DIM * KCODES;         // K
    float* out_sum  = out_cnt + KCODES;                       // D*K
    float* out_loss = out_sum + (size_t)DIM * KCODES;         // 1

    vq_init_kernel<<<(DIM * KCODES + 255) / 256, 256, 0, stream>>>(
        emb, ET, counts, sums, lossAcc);
    vq_enorm_kernel<<<(KCODES + 255) / 256, 256, 0, stream>>>(ET, enorm);
    const int tiles = N_ROWS / 16;            // 8192, 8 waves/block
    vq_main_kernel<<<tiles / 8, 256, 0, stream>>>(
        x, ET, enorm, out_q, counts, sums, lossAcc);
    vq_finalize_kernel<<<(DIM * KCODES + 255) / 256, 256, 0, stream>>>(
        ema_cnt, emb_sum, counts, sums, lossAcc,
        out_emb, out_cnt, out_sum, out_loss);
}